// Semantic_AlignmentModel_36756330119362
// MI455X (gfx1250) — compile-verified
//
#include <hip/hip_runtime.h>

// ---------------------------------------------------------------------------
// Problem constants (from reference): B=4, Lt=128, Lv=196, D=768.
// ---------------------------------------------------------------------------
#define BATCH 4
#define LT    128
#define LV    196
#define DIM   768

typedef __attribute__((ext_vector_type(2))) float v2f;
typedef __attribute__((ext_vector_type(8))) float v8f;

// gfx1250 hardware tanh (transcendental pipe) — 77M evaluations dominate this
// workload. Pure asm (no volatile) so the scheduler can interleave it.
__device__ __forceinline__ float hw_tanh(float x) {
    float y;
    asm("v_tanh_f32 %0, %1" : "=v"(y) : "v"(x));
    return y;
}

// ---------------------------------------------------------------------------
// Kernel 1: projection GEMM  C[b] = A[b] @ W   (TRANSW=0)
//                            C[b] = A[b] @ W^T (TRANSW=1)
// TRANSW is a template parameter so the K-loop body is branch-free
// straight-line code (round-2 disasm showed the runtime-uniform branch being
// re-evaluated with SCC/exec gymnastics every iteration).
//
// One wave computes a 16x64 tile (4 accumulators) via V_WMMA_F32_16X16X4_F32;
// each A fragment is loaded once (global_load_b64) and reused across 4 WMMAs.
// Ragged rows (Lv=196): row index is CLAMPED (not branch-guarded) so loads
// stay unconditional and EXEC stays all-1s.
//
// ISA layouts (§7.12.2), wave32:
//   A (16x4 f32, 2 VGPRs): h = lane>>4, m = lane&15; vgpr j = A[m][k + 2h + j]
//   B (4x16 f32, 2 VGPRs): vgpr j = B[k + 2h + j][n = lane&15]
//   C/D (16x16 f32, 8 VGPRs): vgpr r = C[r + 8h][lane&15]
// ---------------------------------------------------------------------------
#define NTILE 4   // N-tiles per wave -> 16x64 output per wave

template <int TRANSW>
__global__ __launch_bounds__(32)
void proj_wmma(const float* __restrict__ A, const float* __restrict__ W,
               float* __restrict__ C, int rows)
{
    const int lane = threadIdx.x;
    const int half = lane >> 4;      // 0 or 1
    const int l16  = lane & 15;
    const int m0   = blockIdx.x * 16;
    const int n0   = blockIdx.y * 64;
    const int b    = blockIdx.z;

    const float* Ab = A + (size_t)b * rows * DIM;
    float*       Cb = C + (size_t)b * rows * DIM;

    const int arow = min(m0 + l16, rows - 1);         // clamp, no divergence
    const float* Aptr = Ab + (size_t)arow * DIM + 2 * half;

    // Running B pointers (one per N-tile), advanced by the K stride.
    const float* Bptr[NTILE];
#pragma unroll
    for (int nt = 0; nt < NTILE; ++nt) {
        const int ncol = n0 + nt * 16 + l16;
        Bptr[nt] = TRANSW ? (W + (size_t)ncol * DIM + 2 * half)   // W[n][k]
                          : (W + (size_t)(2 * half) * DIM + ncol); // W[k][n]
    }

    v8f acc[NTILE] = {};
#pragma unroll 2
    for (int k = 0; k < DIM; k += 4) {
        const v2f afrag = *(const v2f*)(Aptr + k);    // global_load_b64
#pragma unroll
        for (int nt = 0; nt < NTILE; ++nt) {
            v2f bfrag;
            if (TRANSW) {                              // contiguous pair
                bfrag = *(const v2f*)(Bptr[nt] + k);
            } else {                                   // DIM-strided pair
                bfrag.x = Bptr[nt][(size_t)k * DIM];
                bfrag.y = Bptr[nt][(size_t)(k + 1) * DIM];
            }
            acc[nt] = __builtin_amdgcn_wmma_f32_16x16x4_f32(
                false, afrag, false, bfrag, (short)0, acc[nt], false, false);
        }
    }
#pragma unroll
    for (int nt = 0; nt < NTILE; ++nt) {
#pragma unroll
        for (int r = 0; r < 8; ++r) {
            const int m = m0 + r + 8 * half;
            if (m < rows)
                Cb[(size_t)m * DIM + n0 + nt * 16 + l16] = acc[nt][r];
        }
    }
}

// ---------------------------------------------------------------------------
// Kernel 2: affinity  Caff[b,t,v] = tanh( sum_k T1[b,t,k] * V1[b,v,k] )
// GEMM with B-side = V1^T (per-lane contiguous b64 gathers), K = DIM,
// 16t x 64v tile per wave, fused hardware tanh on the accumulators.
// ---------------------------------------------------------------------------
__global__ __launch_bounds__(32)
void affinity_wmma(const float* __restrict__ T1, const float* __restrict__ V1,
                   float* __restrict__ Caff)
{
    const int lane = threadIdx.x;
    const int half = lane >> 4;
    const int l16  = lane & 15;
    const int t0   = blockIdx.x * 16;   // 8 tiles
    const int v0   = blockIdx.y * 64;   // 4 tiles (last ragged)
    const int b    = blockIdx.z;

    const float* T1p = T1 + ((size_t)b * LT + t0 + l16) * DIM + 2 * half;
    const float* V1b = V1 + (size_t)b * LV * DIM;
    float*       Cb  = Caff + (size_t)b * LT * LV;

    const float* Vptr[NTILE];
#pragma unroll
    for (int nt = 0; nt < NTILE; ++nt) {
        const int vc = min(v0 + nt * 16 + l16, LV - 1);   // clamp
        Vptr[nt] = V1b + (size_t)vc * DIM + 2 * half;
    }

    v8f acc[NTILE] = {};
#pragma unroll 2
    for (int k = 0; k < DIM; k += 4) {
        const v2f afrag = *(const v2f*)(T1p + k);
#pragma unroll
        for (int nt = 0; nt < NTILE; ++nt) {
            const v2f bfrag = *(const v2f*)(Vptr[nt] + k);
            acc[nt] = __builtin_amdgcn_wmma_f32_16x16x4_f32(
                false, afrag, false, bfrag, (short)0, acc[nt], false, false);
        }
    }
#pragma unroll
    for (int nt = 0; nt < NTILE; ++nt) {
        const int v = v0 + nt * 16 + l16;
#pragma unroll
        for (int r = 0; r < 8; ++r) {
            const int t = t0 + r + 8 * half;
            if (v < LV)
                Cb[(size_t)t * LV + v] = hw_tanh(acc[nt][r]);
        }
    }
}

// ---------------------------------------------------------------------------
// Kernel 3 (dominant stage):
//   score[b,t,v] = sum_d text[b,t,d] * tanh(T2[b,t,d] + V3[b,v,d]*Caff[b,t,v])
// Never materializes M (308 MB of traffic avoided). 256 threads per
// 16t x 16v tile; d staged through LDS in 64-wide chunks:
//   s_tt : interleaved {text, T2} float2 (t-indexed -> broadcast reads)
//   s_V3 : stride 66 so paired float2 reads across 16 v-lanes are
//          bank-conflict-free (banks 2v, 2v+1)
// Inner step handles 2 d: ds_b64 + broadcast b128 + 2x(fma, v_tanh, fma).
// ---------------------------------------------------------------------------
#define TT 16
#define TV 16
#define DC 64

__global__ __launch_bounds__(256)
void score_kernel(const float* __restrict__ text, const float* __restrict__ T2,
                  const float* __restrict__ V3, const float* __restrict__ Caff,
                  float* __restrict__ out)
{
    __shared__ float2 s_tt[TT][DC];        // {text, T2}
    __shared__ float  s_V3[TV][DC + 2];    // stride 66 floats

    const int b   = blockIdx.z;
    const int t0  = blockIdx.x * TT;
    const int v0  = blockIdx.y * TV;
    const int tid = threadIdx.x;
    const int vl  = tid & 15;       // v within tile
    const int tl  = tid >> 4;       // t within tile
    const int t   = t0 + tl;
    const int v   = v0 + vl;
    const bool v_ok = (v < LV);

    const float* textb = text + ((size_t)b * LT + t0) * DIM;
    const float* T2b   = T2   + ((size_t)b * LT + t0) * DIM;
    const float* V3b   = V3   + (size_t)b * LV * DIM;

    const float cval = Caff[((size_t)b * LT + t) * LV + min(v, LV - 1)];
    float acc = 0.0f;

    for (int d0 = 0; d0 < DIM; d0 += DC) {
        __syncthreads();
        // 256 threads cooperatively stage 16x64 panels (4 elems per thread)
        for (int i = tid; i < TT * DC; i += 256) {
            const int r = i / DC, c = i % DC;
            s_tt[r][c] = make_float2(textb[(size_t)r * DIM + d0 + c],
                                     T2b  [(size_t)r * DIM + d0 + c]);
            const int vr = min(v0 + r, LV - 1);           // clamp, no branch
            s_V3[r][c]   = V3b[(size_t)vr * DIM + d0 + c];
        }
        __syncthreads();
#pragma unroll
        for (int c = 0; c < DC; c += 2) {
            const float2 vv  = *(const float2*)&s_V3[vl][c];
            const float2 tt0 = s_tt[tl][c];
            const float2 tt1 = s_tt[tl][c + 1];
            acc = fmaf(tt0.x, hw_tanh(fmaf(vv.x, cval, tt0.y)), acc);
            acc = fmaf(tt1.x, hw_tanh(fmaf(vv.y, cval, tt1.y)), acc);
        }
    }
    if (v_ok)
        out[((size_t)b * LT + t) * LV + v] = acc;
}

// ---------------------------------------------------------------------------
// Launch: 4 projection GEMMs -> affinity GEMM(+tanh) -> fused score stage.
// Workspace layout (floats): T1 | T2 | V1 | V3 | Caff  (~8.4 MB total).
// ---------------------------------------------------------------------------
extern "C" void kernel_launch(void* const* d_in, const int* in_sizes, int n_in,
                              void* d_out, int out_size, void* d_ws, size_t ws_size,
                              hipStream_t stream) {
    (void)in_sizes; (void)n_in; (void)out_size; (void)ws_size;

    const float* text = (const float*)d_in[0];   // [4,128,768]
    const float* vis  = (const float*)d_in[1];   // [4,196,768]
    const float* W1   = (const float*)d_in[2];   // [768,768]
    const float* W2   = (const float*)d_in[3];
    const float* W3   = (const float*)d_in[4];
    float* out = (float*)d_out;                  // [4,128,196]

    float* ws = (float*)d_ws;
    const size_t nT = (size_t)BATCH * LT * DIM;  // 393216
    const size_t nV = (size_t)BATCH * LV * DIM;  // 602112
    float* T1   = ws;
    float* T2   = T1 + nT;
    float* V1   = T2 + nT;
    float* V3   = V1 + nV;
    float* Caff = V3 + nV;                       // 4*128*196

    const dim3 wave(32);
    const int mTilesT = LT / 16;                 // 8
    const int mTilesV = (LV + 15) / 16;          // 13
    const int nTiles64 = DIM / 64;               // 12
    const int vTiles64 = (LV + 63) / 64;         // 4

    // Projections (fp32 WMMA, full fidelity; GEMM FLOPs are negligible here)
    proj_wmma<0><<<dim3(mTilesT, nTiles64, BATCH), wave, 0, stream>>>(text, W1, T1, LT);
    proj_wmma<0><<<dim3(mTilesT, nTiles64, BATCH), wave, 0, stream>>>(text, W2, T2, LT);
    proj_wmma<1><<<dim3(mTilesV, nTiles64, BATCH), wave, 0, stream>>>(vis,  W1, V1, LV);
    proj_wmma<1><<<dim3(mTilesV, nTiles64, BATCH), wave, 0, stream>>>(vis,  W3, V3, LV);

    // Co-attention affinity with fused hardware tanh
    affinity_wmma<<<dim3(mTilesT, vTiles64, BATCH), wave, 0, stream>>>(T1, V1, Caff);

    // Fused M/score stage (the 77M-tanh hot loop)
    score_kernel<<<dim3(mTilesT, mTilesV, BATCH), dim3(256), 0, stream>>>(
        text, T2, V3, Caff, out);
}